// FeatureDistillation_24653112279710
// MI455X (gfx1250) — compile-verified
//
#include <hip/hip_runtime.h>
#include <cmath>

typedef __attribute__((ext_vector_type(2))) float v2f;
typedef __attribute__((ext_vector_type(8))) float v8f;

#define IN_H 299
#define UP   304
#define PLANE (UP*UP)               /* 92416 */
#define GROUPS_PER_PLANE 361        /* 92416 / 256 */
#define NPLANES (128*3)
#define TOTAL_GROUPS (NPLANES*GROUPS_PER_PLANE)   /* 138624 */
#define WAVES_PER_BLOCK 8
#define TILE_STRIDE 17              /* gcd(17,64)=1 -> conflict-free column reads */

struct DctConsts { float d[64]; };  // D[k][n], orthonormal 8-pt DCT-II

// ---------------------------------------------------------------------------
// Kernel 1: bilinear 299->304 + range map, then per 8x8 raster block:
//           DCT -> quant/dequant -> IDCT via chained V_WMMA_F32_16X16X4_F32.
// One wave handles one group of 4 blocks (a 16x16 tile, 256 plane elements).
// ---------------------------------------------------------------------------
__global__ void __launch_bounds__(256)
dct_quant_kernel(const float* __restrict__ x, float* __restrict__ ws, DctConsts K)
{
    const int lane = threadIdx.x & 31;
    const int wave = threadIdx.x >> 5;
    const int group = blockIdx.x * WAVES_PER_BLOCK + wave;   // grid sized exactly
    const int plane = group / GROUPS_PER_PLANE;
    const int g     = group - plane * GROUPS_PER_PLANE;
    const int b     = plane / 3;
    const int ch    = plane - b * 3;
    const int half  = lane >> 4;
    const int lm    = lane & 15;

    __shared__ float lds[WAVES_PER_BLOCK][16 * TILE_STRIDE];
    float* tile = lds[wave];

    // Constant operands. A-layout(M): lane L, step s, reg v -> M[L%16][4s+v+2*(L/16)].
    // Identity: A-layout(M) == B-layout(M^T). So:
    //   CA = A-layout(blockdiag(D))   : stage-1 A and stage-2 B (== blockdiag(D^T) as B)
    //   CB = A-layout(blockdiag(D^T)) : stage-3 A and stage-4 B (== blockdiag(D)   as B)
    float CA[4][2], CB[4][2];
#pragma unroll
    for (int s = 0; s < 4; ++s)
#pragma unroll
        for (int v = 0; v < 2; ++v) {
            int r = lm, c = 4 * s + v + 2 * half;
            bool diag = (r < 8) == (c < 8);
            int rb = r & 7, cb = c & 7;
            CA[s][v] = diag ? K.d[rb * 8 + cb] : 0.0f;
            CB[s][v] = diag ? K.d[cb * 8 + rb] : 0.0f;
        }

    // ---- Stage 0: compute 256 resized, range-mapped values into the LDS tile
    // Tile layout [[X0,X1],[X2,X3]]: block q at rows (q>>1)*8.., cols (q&1)*8..
    const float scale = (float)IN_H / (float)UP;
    const int base = g * 256;
    const float* __restrict__ xp = x + (size_t)b * IN_H * IN_H * 3 + ch;
#pragma unroll
    for (int t = 0; t < 8; ++t) {
        int pl = t * 32 + lane;          // 0..255
        int q  = pl >> 6;
        int j  = (pl >> 3) & 7;
        int k  = pl & 7;
        int P  = base + pl;              // flat position in 304x304 plane
        int R  = P / UP;
        int c  = P - R * UP;
        float ys = ((float)R + 0.5f) * scale - 0.5f;
        float xs = ((float)c + 0.5f) * scale - 0.5f;
        float yf = floorf(ys), xf = floorf(xs);
        float ty = ys - yf,    tx = xs - xf;
        int y0 = (int)yf, x0 = (int)xf;
        int y0c = min(max(y0,     0), IN_H - 1);
        int y1c = min(max(y0 + 1, 0), IN_H - 1);
        int x0c = min(max(x0,     0), IN_H - 1);
        int x1c = min(max(x0 + 1, 0), IN_H - 1);
        float v00 = xp[((size_t)y0c * IN_H + x0c) * 3];
        float v01 = xp[((size_t)y0c * IN_H + x1c) * 3];
        float v10 = xp[((size_t)y1c * IN_H + x0c) * 3];
        float v11 = xp[((size_t)y1c * IN_H + x1c) * 3];
        float v0  = v00 + tx * (v01 - v00);
        float v1  = v10 + tx * (v11 - v10);
        float val = v0 + ty * (v1 - v0);
        val = (val + 1.0f) * 127.5f;     // [-1,1] -> [0,255]
        int r16 = (q >> 1) * 8 + j;
        int c16 = (q & 1) * 8 + k;
        tile[r16 * TILE_STRIDE + c16] = val;
    }
    // single wave: LDS ops are issue-ordered per wave -> no barrier needed

    auto readB = [&](float out[4][2]) {   // B-layout: B[4s+v+2*half][lm]
#pragma unroll
        for (int s = 0; s < 4; ++s)
#pragma unroll
            for (int v = 0; v < 2; ++v)
                out[s][v] = tile[(4 * s + v + 2 * half) * TILE_STRIDE + lm];
    };
    auto readA = [&](float out[4][2]) {   // A-layout: A[lm][4s+v+2*half]
#pragma unroll
        for (int s = 0; s < 4; ++s)
#pragma unroll
            for (int v = 0; v < 2; ++v)
                out[s][v] = tile[lm * TILE_STRIDE + (4 * s + v + 2 * half)];
    };
    auto writeC = [&](const v8f& creg) {  // C-layout: row j+8*half, col lm
#pragma unroll
        for (int j2 = 0; j2 < 8; ++j2)
            tile[(j2 + 8 * half) * TILE_STRIDE + lm] = creg[j2];
    };
    auto mm16 = [&](const float A[4][2], const float B[4][2]) { // 16x16x16 via 4 k-steps
        v8f acc = {};
#pragma unroll
        for (int s = 0; s < 4; ++s) {
            v2f a = { A[s][0], A[s][1] };
            v2f bb = { B[s][0], B[s][1] };
            acc = __builtin_amdgcn_wmma_f32_16x16x4_f32(
                false, a, false, bb, (short)0, acc, false, false);
        }
        return acc;
    };

    // ---- Stage 1: T1 = blockdiag(D) @ Xtile
    float Bx[4][2]; readB(Bx);
    v8f t1 = mm16(CA, Bx);
    writeC(t1);

    // ---- Stage 2: DCT = T1 @ blockdiag(D^T)
    float At[4][2]; readA(At);
    v8f dct = mm16(At, CA);

    // ---- Quant/dequant in C-registers (round half-to-even = v_rndne)
#pragma unroll
    for (int j2 = 0; j2 < 8; ++j2) {
        int r16 = j2 + 8 * half, c16 = lm;
        int jb = r16 & 7, kb = c16 & 7;
        float qv = (jb < 4 && kb < 4) ? 25.0f : 30.0f;
        dct[j2] = rintf(dct[j2] / qv) * qv;
    }
    writeC(dct);

    // ---- Stage 3: T2 = blockdiag(D^T) @ Qdct
    float Bq[4][2]; readB(Bq);
    v8f t2 = mm16(CB, Bq);
    writeC(t2);

    // ---- Stage 4: REC = T2 @ blockdiag(D)
    float A2[4][2]; readA(A2);
    v8f rec = mm16(A2, CB);

    // ---- Write reconstructed values back to flat plane positions in ws
    float* __restrict__ wp = ws + (size_t)plane * PLANE + base;
#pragma unroll
    for (int j2 = 0; j2 < 8; ++j2) {
        int r16 = j2 + 8 * half, c16 = lm;
        int q  = (r16 >> 3) * 2 + (c16 >> 3);
        int jb = r16 & 7, kb = c16 & 7;
        wp[q * 64 + jb * 8 + kb] = rec[j2];
    }
}

// ---------------------------------------------------------------------------
// Kernel 2: bilinear 304->299, clip, map back to [-1,1]. One thread per
// output pixel (all 3 channels) -> coalesced ws reads and out writes.
// ---------------------------------------------------------------------------
__global__ void __launch_bounds__(256)
downsample_kernel(const float* __restrict__ ws, float* __restrict__ out)
{
    long long i = (long long)blockIdx.x * 256 + threadIdx.x;
    const long long total = 128LL * IN_H * IN_H;
    if (i >= total) return;
    int ox = (int)(i % IN_H);
    long long r = i / IN_H;
    int oy = (int)(r % IN_H);
    int b  = (int)(r / IN_H);

    const float s2 = (float)UP / (float)IN_H;
    float ys = ((float)oy + 0.5f) * s2 - 0.5f;
    float xs = ((float)ox + 0.5f) * s2 - 0.5f;
    float yf = floorf(ys), xf = floorf(xs);
    float ty = ys - yf,    tx = xs - xf;
    int y0 = (int)yf, x0 = (int)xf;
    int y0c = min(max(y0,     0), UP - 1);
    int y1c = min(max(y0 + 1, 0), UP - 1);
    int x0c = min(max(x0,     0), UP - 1);
    int x1c = min(max(x0 + 1, 0), UP - 1);

    float* op = out + (size_t)i * 3;
#pragma unroll
    for (int ch = 0; ch < 3; ++ch) {
        const float* __restrict__ p = ws + (size_t)(b * 3 + ch) * PLANE;
        float v00 = p[y0c * UP + x0c];
        float v01 = p[y0c * UP + x1c];
        float v10 = p[y1c * UP + x0c];
        float v11 = p[y1c * UP + x1c];
        float v0  = v00 + tx * (v01 - v00);
        float v1  = v10 + tx * (v11 - v10);
        float val = v0 + ty * (v1 - v0);
        val = fminf(fmaxf(val * (1.0f / 255.0f), 0.0f), 1.0f);
        op[ch] = val * 2.0f - 1.0f;
    }
}

extern "C" void kernel_launch(void* const* d_in, const int* in_sizes, int n_in,
                              void* d_out, int out_size, void* d_ws, size_t ws_size,
                              hipStream_t stream)
{
    (void)in_sizes; (void)n_in; (void)out_size; (void)ws_size;
    const float* x = (const float*)d_in[0];
    float* out = (float*)d_out;
    float* ws  = (float*)d_ws;   // needs 128*3*304*304*4 = ~142 MB

    DctConsts K;
    const double pi = 3.14159265358979323846;
    for (int k = 0; k < 8; ++k)
        for (int n = 0; n < 8; ++n) {
            double v = std::cos(pi * (2 * n + 1) * k / 16.0) * std::sqrt(2.0 / 8.0);
            if (k == 0) v /= std::sqrt(2.0);
            K.d[k * 8 + n] = (float)v;
        }

    // Kernel 1: exact grid so EXEC is all-ones for every wave (WMMA requirement)
    dct_quant_kernel<<<TOTAL_GROUPS / WAVES_PER_BLOCK, 256, 0, stream>>>(x, ws, K);

    // Kernel 2
    long long total = 128LL * IN_H * IN_H;
    int blocks = (int)((total + 255) / 256);
    downsample_kernel<<<blocks, 256, 0, stream>>>(ws, out);
}